// BCIM_36532991820508
// MI455X (gfx1250) — compile-verified
//
#include <hip/hip_runtime.h>
#include <hip/hip_bf16.h>
#include <math.h>

// ---------------------------------------------------------------------------
// Problem constants (from reference): B=8, S=2048, C=768 -> H=W=32, K=1536.
// Feature vector at spatial p is concat(patch[b,p,:], patch[b,p+1024,:]).
// ---------------------------------------------------------------------------
#define NB      8
#define SEQ     2048
#define CH      768
#define HW      32
#define POS     1024
#define KDIM    1536
#define CHUNK   32      // K-chunk per wave per iteration
#define STRIDE  36      // padded LDS stride per position-row (floats)
#define GSTRIDE 33      // padded Gram stride
#define NCHUNKS 12      // 1536 / (CHUNK * 4 waves)

typedef __attribute__((ext_vector_type(2))) float v2f;
typedef __attribute__((ext_vector_type(8))) float v8f;

__device__ __forceinline__ v8f wmma_f32_16x16x4(v2f a, v2f b, v8f c) {
    // D = A(16x4,f32) * B(4x16,f32) + C(16x16,f32)
    return __builtin_amdgcn_wmma_f32_16x16x4_f32(
        /*neg_a=*/false, a, /*neg_b=*/false, b,
        /*c_mod=*/(short)0, c, /*reuse_a=*/false, /*reuse_b=*/false);
}

// ---------------------------------------------------------------------------
// Kernel 1: per (batch, row h): Gram bands via V_WMMA_F32_16X16X4_F32.
//   G_hh  = X_h   X_h^T   -> nn (diag), dE (super-diag)
//   G_h,h+1 = X_h X_{h+1}^T -> dS (diag), dSE (super-diag), dSW (sub-diag)
// 4 waves split K=1536; partials summed deterministically by wave 0.
// ---------------------------------------------------------------------------
__global__ __launch_bounds__(128) void bcim_gram_kernel(
    const float* __restrict__ in,
    float* __restrict__ wsNN, float* __restrict__ wsE,
    float* __restrict__ wsS,  float* __restrict__ wsSE,
    float* __restrict__ wsSW)
{
    __shared__ float stage[4][2 * HW * STRIDE];   // per-wave private staging
    __shared__ float band[4][5][HW];              // per-wave band partials

    const int blk  = blockIdx.x;      // 0..255
    const int b    = blk >> 5;
    const int h    = blk & 31;
    const int wv   = threadIdx.x >> 5;
    const int lane = threadIdx.x & 31;
    const int half = lane >> 4;       // 0: K=0..1 / 1: K=2..3 (fp32 WMMA layout)
    const int l15  = lane & 15;

    float* st = &stage[wv][0];

    v8f aHH00 = {}, aHH01 = {}, aHH10 = {}, aHH11 = {};
    v8f aX00  = {}, aX01  = {}, aX10  = {}, aX11  = {};

    for (int cp = 0; cp < NCHUNKS; ++cp) {
        const int f0 = (cp * 4 + wv) * CHUNK;       // this wave's K-offset
        const int kk = (f0 >= CH) ? 1 : 0;          // which token half
        const int c0 = f0 - kk * CH;

        // --- stage 2 rows x 32 positions x 32 channels into LDS (float4) ---
        for (int i = 0; i < 16; ++i) {
            const int flat = i * 32 + lane;
            const int pr = flat >> 3;               // 0..63 position-row
            const int j  = flat & 7;                // float4 within chunk
            const int r  = pr >> 5;                 // 0: row h, 1: row h+1
            const int w  = pr & 31;
            const int rr = (h + r > 31) ? 31 : (h + r);   // clamp (h=31 cross unused)
            const int s  = kk * 1024 + rr * HW + w;
            const float4 v = *(const float4*)(in + ((size_t)(b * SEQ + s)) * CH + c0 + j * 4);
            *(float4*)(st + pr * STRIDE + j * 4) = v;
        }
        // --- prefetch next K-chunk (CDNA5 global_prefetch_b8) ---
        if (cp + 1 < NCHUNKS) {
            const int fn  = ((cp + 1) * 4 + wv) * CHUNK;
            const int kkn = (fn >= CH) ? 1 : 0;
            const int c0n = fn - kkn * CH;
            const int pr  = lane * 2;
            const int rr  = (h + (pr >> 5) > 31) ? 31 : (h + (pr >> 5));
            const int s   = kkn * 1024 + rr * HW + (pr & 31);
            __builtin_prefetch(in + ((size_t)(b * SEQ + s)) * CH + c0n, 0, 0);
        }
        __syncthreads();

        // --- consume: 8 K-steps of 4, 8 WMMAs each ---
        for (int ks = 0; ks < CHUNK / 4; ++ks) {
            const int kloc = ks * 4 + half * 2;
            // fp32 WMMA fragment layout is symmetric for A and B:
            // lane(l15,half) holds element (row l15, K = kloc..kloc+1)
            const v2f a0  = *(const v2f*)(st + (0 * HW + 0  + l15) * STRIDE + kloc);
            const v2f a1  = *(const v2f*)(st + (0 * HW + 16 + l15) * STRIDE + kloc);
            const v2f b10 = *(const v2f*)(st + (1 * HW + 0  + l15) * STRIDE + kloc);
            const v2f b11 = *(const v2f*)(st + (1 * HW + 16 + l15) * STRIDE + kloc);

            aHH00 = wmma_f32_16x16x4(a0, a0,  aHH00);
            aHH01 = wmma_f32_16x16x4(a0, a1,  aHH01);
            aHH10 = wmma_f32_16x16x4(a1, a0,  aHH10);
            aHH11 = wmma_f32_16x16x4(a1, a1,  aHH11);
            aX00  = wmma_f32_16x16x4(a0, b10, aX00);
            aX01  = wmma_f32_16x16x4(a0, b11, aX01);
            aX10  = wmma_f32_16x16x4(a1, b10, aX10);
            aX11  = wmma_f32_16x16x4(a1, b11, aX11);
        }
        __syncthreads();
    }

    // --- dump Gram tiles into this wave's private LDS region ---
    float* GH = st;                     // 32x33
    float* GX = st + HW * GSTRIDE;      // 32x33
    #pragma unroll
    for (int v = 0; v < 8; ++v) {
        const int M0 = half * 8 + v;    // C/D layout: vgpr v, lane half -> M
        GH[(M0     ) * GSTRIDE + (l15     )] = aHH00[v];
        GH[(M0     ) * GSTRIDE + (l15 + 16)] = aHH01[v];
        GH[(M0 + 16) * GSTRIDE + (l15     )] = aHH10[v];
        GH[(M0 + 16) * GSTRIDE + (l15 + 16)] = aHH11[v];
        GX[(M0     ) * GSTRIDE + (l15     )] = aX00[v];
        GX[(M0     ) * GSTRIDE + (l15 + 16)] = aX01[v];
        GX[(M0 + 16) * GSTRIDE + (l15     )] = aX10[v];
        GX[(M0 + 16) * GSTRIDE + (l15 + 16)] = aX11[v];
    }
    __syncthreads();

    {   // per-wave band partials (raw dot sums; normalization happens later)
        const int w = lane;
        band[wv][0][w] = GH[w * GSTRIDE + w];
        band[wv][1][w] = (w < 31) ? GH[w * GSTRIDE + w + 1] : 0.f;
        band[wv][2][w] = GX[w * GSTRIDE + w];
        band[wv][3][w] = (w < 31) ? GX[w * GSTRIDE + w + 1] : 0.f;
        band[wv][4][w] = (w > 0)  ? GX[w * GSTRIDE + w - 1] : 0.f;
    }
    __syncthreads();

    if (wv == 0) {   // deterministic fixed-order partial sum
        const int w = lane;
        const int idx = (b * HW + h) * HW + w;
        wsNN[idx] = band[0][0][w] + band[1][0][w] + band[2][0][w] + band[3][0][w];
        wsE [idx] = band[0][1][w] + band[1][1][w] + band[2][1][w] + band[3][1][w];
        wsS [idx] = band[0][2][w] + band[1][2][w] + band[2][2][w] + band[3][2][w];
        wsSE[idx] = band[0][3][w] + band[1][3][w] + band[2][3][w] + band[3][3][w];
        wsSW[idx] = band[0][4][w] + band[1][4][w] + band[2][4][w] + band[3][4][w];
    }
}

// ---------------------------------------------------------------------------
// Kernel 2: combine bands (with symmetry + boundary guards) into sim = acc/81.
// ---------------------------------------------------------------------------
__device__ __forceinline__ float nrm_of(const float* nn, int q) {
    return fmaxf(sqrtf(nn[q]), 1e-8f);
}

__global__ __launch_bounds__(256) void bcim_sim_kernel(
    const float* __restrict__ NN, const float* __restrict__ dE,
    const float* __restrict__ dS, const float* __restrict__ dSE,
    const float* __restrict__ dSW, float* __restrict__ SIM)
{
    const int t = blockIdx.x * 256 + threadIdx.x;   // 0..8191 = b*1024 + p
    const int p = t & 1023;
    const int h = p >> 5;
    const int w = p & 31;

    const float nn = NN[t];
    const float n  = fmaxf(sqrtf(nn), 1e-8f);
    float acc = nn / (n * n);                                 // self term
    if (w < 31)           acc += dE [t]      / (n * nrm_of(NN, t + 1));   // E
    if (w > 0)            acc += dE [t - 1]  / (n * nrm_of(NN, t - 1));   // W
    if (h < 31)           acc += dS [t]      / (n * nrm_of(NN, t + 32));  // S
    if (h > 0)            acc += dS [t - 32] / (n * nrm_of(NN, t - 32));  // N
    if (h < 31 && w < 31) acc += dSE[t]      / (n * nrm_of(NN, t + 33));  // SE
    if (h > 0  && w > 0)  acc += dSE[t - 33] / (n * nrm_of(NN, t - 33));  // NW
    if (h < 31 && w > 0)  acc += dSW[t]      / (n * nrm_of(NN, t + 31));  // SW
    if (h > 0  && w < 31) acc += dSW[t - 31] / (n * nrm_of(NN, t - 31));  // NE

    SIM[t] = acc * (1.0f / 81.0f);
}

// ---------------------------------------------------------------------------
// Kernel 3: out[b,s,c] = in[b,s,c] * sim[b, s mod 1024].  Main HBM traffic.
// One block per token row; float4 fully-coalesced stream.
// ---------------------------------------------------------------------------
__global__ __launch_bounds__(192) void bcim_scale_kernel(
    const float4* __restrict__ in4, const float* __restrict__ SIM,
    float4* __restrict__ out4)
{
    const int bt = blockIdx.x;         // 0..16383 = b*2048 + s
    const int b  = bt >> 11;
    const int s  = bt & 2047;
    const float m = SIM[b * POS + (s & 1023)];
    const size_t base = (size_t)bt * (CH / 4);
    float4 v = in4[base + threadIdx.x];
    v.x *= m; v.y *= m; v.z *= m; v.w *= m;
    out4[base + threadIdx.x] = v;
}

// ---------------------------------------------------------------------------
extern "C" void kernel_launch(void* const* d_in, const int* in_sizes, int n_in,
                              void* d_out, int out_size, void* d_ws, size_t ws_size,
                              hipStream_t stream)
{
    const float* in = (const float*)d_in[0];
    float* out = (float*)d_out;

    float* ws   = (float*)d_ws;
    float* wsNN = ws;                 // 8192
    float* wsE  = ws + 8192;
    float* wsS  = ws + 16384;
    float* wsSE = ws + 24576;
    float* wsSW = ws + 32768;
    float* wsSIM= ws + 40960;

    bcim_gram_kernel<<<NB * HW, 128, 0, stream>>>(in, wsNN, wsE, wsS, wsSE, wsSW);
    bcim_sim_kernel<<<(NB * POS) / 256, 256, 0, stream>>>(wsNN, wsE, wsS, wsSE, wsSW, wsSIM);
    bcim_scale_kernel<<<NB * SEQ, CH / 4, 0, stream>>>((const float4*)in, wsSIM, (float4*)out);
}